// Laplacian_11536282157538
// MI455X (gfx1250) — compile-verified
//
#include <hip/hip_runtime.h>
#include <math.h>

typedef float v2f __attribute__((ext_vector_type(2)));
typedef float v8f __attribute__((ext_vector_type(8)));

#define D_IN   128
#define HIDDEN 512
#define BATCH  512

// CDNA5 has a native transcendental tanh (v_tanh_f32). Use the builtin when
// the toolchain exposes it; otherwise fall back to libm tanhf.
#if defined(__has_builtin)
#if __has_builtin(__builtin_amdgcn_tanhf)
#define FAST_TANH(x) __builtin_amdgcn_tanhf(x)
#endif
#endif
#ifndef FAST_TANH
#define FAST_TANH(x) tanhf(x)
#endif

// ---------------------------------------------------------------------------
// Kernel 0: s[j] = sum_i W1[i,j]^2   (Hessian-trace collapse weights)
// ---------------------------------------------------------------------------
__global__ void lap_colsumsq_kernel(const float* __restrict__ W1,
                                    float* __restrict__ s) {
  int j = blockIdx.x * blockDim.x + threadIdx.x;
  if (j >= HIDDEN) return;
  float acc = 0.f;
#pragma unroll 8
  for (int i = 0; i < D_IN; ++i) {
    float w = W1[i * HIDDEN + j];
    acc = fmaf(w, w, acc);
  }
  s[j] = acc;
}

// ---------------------------------------------------------------------------
// Kernel 1: z = x@W1 + b1 via V_WMMA_F32_16X16X4_F32, fused tanh (v_tanh_f32).
// Stores hT[j*B+b] = tanh(z) and gT[j*B+b] = (1-h^2)*W2[j]  (transposed so
// gT is directly the B-matrix layout source for GEMM2 and reductions coalesce).
// One wave = one 16x16 tile; 4 waves per block; EXEC all-ones (no divergence).
// ---------------------------------------------------------------------------
__global__ void lap_gemm1_tanh_kernel(const float* __restrict__ x,
                                      const float* __restrict__ W1,
                                      const float* __restrict__ b1,
                                      const float* __restrict__ W2,
                                      float* __restrict__ hT,
                                      float* __restrict__ gT) {
  const int lane = threadIdx.x;           // 0..31
  const int half = lane >> 4;             // 0 or 1
  const int l    = lane & 15;
  const int b0   = blockIdx.x * 16;                              // batch tile (M)
  const int j0   = (blockIdx.y * blockDim.y + threadIdx.y) * 16; // hidden tile (N)

  v8f acc = {};
#pragma unroll 4
  for (int k0 = 0; k0 < D_IN; k0 += 4) {
    const int ka = k0 + 2 * half;
    // A 16x4: lane holds A[m=l, ka], A[m=l, ka+1]  -> contiguous b64 load
    v2f a = *(const v2f*)(x + (b0 + l) * D_IN + ka);
    // B 4x16: lane holds B[ka, n=l], B[ka+1, n=l]
    v2f b;
    b.x = W1[(ka + 0) * HIDDEN + j0 + l];
    b.y = W1[(ka + 1) * HIDDEN + j0 + l];
    acc = __builtin_amdgcn_wmma_f32_16x16x4_f32(
        false, a, false, b, (short)0, acc, false, false);
  }

  const int   j    = j0 + l;      // D-matrix: n = lane&15 (hidden col)
  const float w2   = W2[j];
  const float bias = b1[j];
#pragma unroll
  for (int v = 0; v < 8; ++v) {
    const int b = b0 + v + 8 * half;       // D-matrix: m = v + 8*half (batch row)
    float h  = FAST_TANH(acc[v] + bias);
    float hp = 1.f - h * h;
    hT[j * BATCH + b] = h;
    gT[j * BATCH + b] = hp * w2;
  }
}

// ---------------------------------------------------------------------------
// Kernel 2: F1 = W1 @ gT  (M=128, N=512, K=512) via V_WMMA_F32_16X16X4_F32.
// F1[i,b] = sum_j W1[i,j] * gT[j,b]; writes d_out F1 region [D_IN, BATCH].
// ---------------------------------------------------------------------------
__global__ void lap_gemm2_kernel(const float* __restrict__ W1,
                                 const float* __restrict__ gT,
                                 float* __restrict__ outF1) {
  const int lane = threadIdx.x;
  const int half = lane >> 4;
  const int l    = lane & 15;
  const int i0   = blockIdx.x * 16;                              // D rows (M)
  const int n0   = (blockIdx.y * blockDim.y + threadIdx.y) * 16; // batch cols (N)

  v8f acc = {};
#pragma unroll 4
  for (int k0 = 0; k0 < HIDDEN; k0 += 4) {
    const int ka = k0 + 2 * half;
    // A 16x4 from W1 rows: contiguous pair
    v2f a = *(const v2f*)(W1 + (i0 + l) * HIDDEN + ka);
    // B 4x16 from gT (already [H][B])
    v2f b;
    b.x = gT[(ka + 0) * BATCH + n0 + l];
    b.y = gT[(ka + 1) * BATCH + n0 + l];
    acc = __builtin_amdgcn_wmma_f32_16x16x4_f32(
        false, a, false, b, (short)0, acc, false, false);
  }

#pragma unroll
  for (int v = 0; v < 8; ++v) {
    const int i = i0 + v + 8 * half;
    outF1[i * BATCH + n0 + l] = acc[v];
  }
}

// ---------------------------------------------------------------------------
// Kernel 3: per-batch reductions.
//   f0[b] = sum_j h*W2 + b2          -> replicated into F0[i,b] for all i
//   lap[b]= sum_j (-2 h (1-h^2)) * W2[j] * s[j]
// Reads hT (transposed) so consecutive threads (b) hit consecutive addresses.
// ---------------------------------------------------------------------------
__global__ void lap_reduce_kernel(const float* __restrict__ hT,
                                  const float* __restrict__ W2,
                                  const float* __restrict__ b2,
                                  const float* __restrict__ s,
                                  float* __restrict__ outF0,
                                  float* __restrict__ outLap) {
  int b = blockIdx.x * blockDim.x + threadIdx.x;
  if (b >= BATCH) return;
  float f0 = 0.f, lap = 0.f;
#pragma unroll 4
  for (int j = 0; j < HIDDEN; ++j) {
    float h  = hT[j * BATCH + b];
    float w2 = W2[j];
    float hp = 1.f - h * h;
    f0  = fmaf(h, w2, f0);
    lap = fmaf(-2.f * h * hp * w2, s[j], lap);
  }
  f0 += b2[0];
  outLap[b] = lap;
#pragma unroll 8
  for (int i = 0; i < D_IN; ++i) outF0[i * BATCH + b] = f0;
}

// ---------------------------------------------------------------------------
extern "C" void kernel_launch(void* const* d_in, const int* in_sizes, int n_in,
                              void* d_out, int out_size, void* d_ws, size_t ws_size,
                              hipStream_t stream) {
  const float* x  = (const float*)d_in[0];   // [BATCH, D_IN]
  const float* W1 = (const float*)d_in[1];   // [D_IN, HIDDEN]
  const float* b1 = (const float*)d_in[2];   // [HIDDEN]
  const float* W2 = (const float*)d_in[3];   // [HIDDEN, 1]
  const float* b2 = (const float*)d_in[4];   // [1]

  float* out    = (float*)d_out;
  float* outF0  = out;                        // [D_IN, BATCH]  = 65536
  float* outF1  = out + D_IN * BATCH;         // [D_IN, BATCH]  = 65536
  float* outLap = out + 2 * D_IN * BATCH;     // [BATCH]        = 512

  float* hT = (float*)d_ws;                   // [HIDDEN, BATCH] 1 MB
  float* gT = hT + HIDDEN * BATCH;            // [HIDDEN, BATCH] 1 MB
  float* s  = gT + HIDDEN * BATCH;            // [HIDDEN]        2 KB

  // 0) column sum-of-squares of W1
  lap_colsumsq_kernel<<<(HIDDEN + 255) / 256, 256, 0, stream>>>(W1, s);

  // 1) h = tanh(x@W1+b1); store hT, gT   (32x32 tiles, 4 waves/block)
  lap_gemm1_tanh_kernel<<<dim3(BATCH / 16, HIDDEN / 16 / 4), dim3(32, 4), 0, stream>>>(
      x, W1, b1, W2, hT, gT);

  // 2) F1 = W1 @ gT    (8x32 tiles, 4 waves/block)
  lap_gemm2_kernel<<<dim3(D_IN / 16, BATCH / 16 / 4), dim3(32, 4), 0, stream>>>(
      W1, gT, outF1);

  // 3) F0 (replicated) and Laplacian
  lap_reduce_kernel<<<BATCH / 256, 256, 0, stream>>>(hT, W2, b2, s, outF0, outLap);
}